// PointNetLayer_37718402793769
// MI455X (gfx1250) — compile-verified
//
#include <hip/hip_runtime.h>
#include <hip/hip_bf16.h>

typedef float v2f __attribute__((ext_vector_type(2)));
typedef float v8f __attribute__((ext_vector_type(8)));

#define B_   16
#define N_   4096
#define M_   1024
#define K_   32
#define NCOL (B_ * M_ * K_)   // 524288 columns
#define BN_EPS 1e-5f

__device__ __forceinline__ v8f wmma4(v2f a, v2f b, v8f c) {
  // D = A(16x4, f32) * B(4x16, f32) + C(16x16, f32)
  return __builtin_amdgcn_wmma_f32_16x16x4_f32(
      /*neg_a=*/false, a, /*neg_b=*/false, b,
      /*c_mod=*/(short)0, c, /*reuse_a=*/false, /*reuse_b=*/false);
}

// ---------------- features [B,C,N] -> featT [B,N,C] ----------------
__global__ void feat_transpose_kernel(const float* __restrict__ feat,
                                      float* __restrict__ featT) {
  __shared__ float tile[64][65];
  int b  = blockIdx.x >> 6;          // 64 n-tiles per batch
  int n0 = (blockIdx.x & 63) << 6;   // 64 points per tile
  int t  = threadIdx.x;
  #pragma unroll
  for (int p = 0; p < 16; ++p) {
    int cc = p * 4 + (t >> 6);
    int nn = t & 63;
    tile[nn][cc] = feat[(size_t)(b * 64 + cc) * N_ + n0 + nn]; // coalesced in n
  }
  __syncthreads();
  #pragma unroll
  for (int p = 0; p < 16; ++p) {
    int nn = p * 4 + (t >> 6);
    int cc = t & 63;
    featT[(size_t)(b * N_ + n0 + nn) * 64 + cc] = tile[nn][cc]; // coalesced in c
  }
}

// ---------------- KNN: one block per query ----------------
__global__ void knn_kernel(const float* __restrict__ loc,
                           const float* __restrict__ new_loc,
                           int* __restrict__ idxOut) {
  __shared__ float dist[N_];
  __shared__ float rv[256];
  __shared__ int   ri[256];
  int q = blockIdx.x;            // b*M + m
  int b = q / M_;
  int t = threadIdx.x;
  float qx = new_loc[q * 3 + 0];
  float qy = new_loc[q * 3 + 1];
  float qz = new_loc[q * 3 + 2];
  for (int i = t; i < N_; i += 256) {
    float dx = loc[(size_t)(b * N_ + i) * 3 + 0] - qx;
    float dy = loc[(size_t)(b * N_ + i) * 3 + 1] - qy;
    float dz = loc[(size_t)(b * N_ + i) * 3 + 2] - qz;
    dist[i] = dx * dx + dy * dy + dz * dz;
  }
  __syncthreads();
  for (int r = 0; r < K_; ++r) {
    float best = 3.4e38f; int bi = N_;
    for (int i = t; i < N_; i += 256) {
      float d = dist[i];
      if (d < best) { best = d; bi = i; }   // first (lowest i) wins ties
    }
    rv[t] = best; ri[t] = bi;
    __syncthreads();
    if (t < 32) {
      float bv = rv[t]; int bj = ri[t];
      #pragma unroll
      for (int j = 1; j < 8; ++j) {
        float v = rv[t + 32 * j]; int k = ri[t + 32 * j];
        if (v < bv || (v == bv && k < bj)) { bv = v; bj = k; }
      }
      #pragma unroll
      for (int mask = 16; mask >= 1; mask >>= 1) {
        float v = __shfl_xor(bv, mask);
        int   k = __shfl_xor(bj, mask);
        if (v < bv || (v == bv && k < bj)) { bv = v; bj = k; }
      }
      if (t == 0) { idxOut[q * K_ + r] = bj; dist[bj] = 3.4e38f; }
    }
    __syncthreads();
  }
}

// ---------------- shared epilogue helpers ----------------
__device__ __forceinline__ void stats_atomic(float y, int ch, int row,
                                             float* s_sum, float* s_sq) {
  float s = y, sq = y * y;
  #pragma unroll
  for (int mask = 1; mask <= 8; mask <<= 1) {  // reduce within 16-lane halves
    s  += __shfl_xor(s,  mask);
    sq += __shfl_xor(sq, mask);
  }
  if (row == 0) {            // lanes 0 and 16 -> two distinct channels
    atomicAdd(&s_sum[ch], s);
    atomicAdd(&s_sq[ch],  sq);
  }
}

// ---------------- conv1: gather + 64x67 GEMM + BN1 stats ----------------
__global__ void conv1_kernel(const int* __restrict__ idx,
                             const float* __restrict__ featT,
                             const float* __restrict__ loc,
                             const float* __restrict__ new_loc,
                             const float* __restrict__ W1,   // [64 x 67]
                             float* __restrict__ Y1,         // [64 x NCOL]
                             float* __restrict__ s_sum, float* __restrict__ s_sq) {
  __shared__ float X[68 * 32];     // 68 input rows (3 xyz + 64 feat + 1 pad)
  int q = blockIdx.x;
  int b = q / M_;
  int t = threadIdx.x;
  int col = t & 31, part = t >> 5;
  int n = idx[q * K_ + col];
  float qx = new_loc[q * 3 + 0];
  float qy = new_loc[q * 3 + 1];
  float qz = new_loc[q * 3 + 2];
  const float* fp = featT + (size_t)(b * N_ + n) * 64;
  const float* lp = loc   + (size_t)(b * N_ + n) * 3;
  for (int r = part; r < 68; r += 8) {
    float v;
    if      (r == 0) v = lp[0] - qx;
    else if (r == 1) v = lp[1] - qy;
    else if (r == 2) v = lp[2] - qz;
    else if (r < 67) v = fp[r - 3];
    else             v = 0.f;
    X[r * 32 + col] = v;
  }
  __syncthreads();
  int lane = t & 31, wave = t >> 5;
  int mt = wave >> 1, nt = wave & 1;
  int row = lane & 15, hi = lane >> 4;
  v8f acc = {0.f, 0.f, 0.f, 0.f, 0.f, 0.f, 0.f, 0.f};
  #pragma unroll
  for (int kk = 0; kk < 17; ++kk) {
    int c0 = kk * 4 + hi * 2;
    int wr = mt * 16 + row;
    v2f a, bf;
    a.x = (c0     < 67) ? W1[wr * 67 + c0]     : 0.f;
    a.y = (c0 + 1 < 67) ? W1[wr * 67 + c0 + 1] : 0.f;
    bf.x = X[c0 * 32 + nt * 16 + row];
    bf.y = X[(c0 + 1) * 32 + nt * 16 + row];
    acc = wmma4(a, bf, acc);
  }
  int colg = q * K_ + nt * 16 + row;
  #pragma unroll
  for (int v = 0; v < 8; ++v) {
    int ch = mt * 16 + v + hi * 8;
    float y = acc[v];
    Y1[(size_t)ch * NCOL + colg] = y;
    stats_atomic(y, ch, row, s_sum, s_sq);
  }
}

// ---------------- finalize BN: stats -> (a, b') ----------------
__global__ void finalize_bn(const float* __restrict__ sum, const float* __restrict__ sq,
                            const float* __restrict__ g, const float* __restrict__ bias,
                            float* __restrict__ a, float* __restrict__ bp, int nch) {
  int t = threadIdx.x;
  if (t < nch) {
    const float inv = 1.0f / (float)NCOL;
    float mean = sum[t] * inv;
    float var  = sq[t] * inv - mean * mean;   // biased variance, as torch BN
    float rs   = rsqrtf(var + BN_EPS);
    float aa   = g[t] * rs;
    a[t]  = aa;
    bp[t] = bias[t] - mean * aa;
  }
}

// ---------------- conv2: BN1+ReLU fused load, 64x64 GEMM, BN2 stats ----------------
__global__ void conv2_kernel(const float* __restrict__ Y1,
                             const float* __restrict__ W2,   // [64 x 64]
                             const float* __restrict__ a1, const float* __restrict__ b1p,
                             float* __restrict__ Y2,
                             float* __restrict__ s_sum, float* __restrict__ s_sq) {
  __shared__ float X[64 * 32];
  int q = blockIdx.x;
  int t = threadIdx.x;
  int col = t & 31, part = t >> 5;
  for (int r = part; r < 64; r += 8) {
    float y = Y1[(size_t)r * NCOL + q * K_ + col];
    X[r * 32 + col] = fmaxf(a1[r] * y + b1p[r], 0.f);
  }
  __syncthreads();
  int lane = t & 31, wave = t >> 5;
  int mt = wave >> 1, nt = wave & 1;
  int row = lane & 15, hi = lane >> 4;
  v8f acc = {0.f, 0.f, 0.f, 0.f, 0.f, 0.f, 0.f, 0.f};
  #pragma unroll
  for (int kk = 0; kk < 16; ++kk) {
    int c0 = kk * 4 + hi * 2;
    int wr = mt * 16 + row;
    v2f a, bf;
    a.x = W2[wr * 64 + c0];
    a.y = W2[wr * 64 + c0 + 1];
    bf.x = X[c0 * 32 + nt * 16 + row];
    bf.y = X[(c0 + 1) * 32 + nt * 16 + row];
    acc = wmma4(a, bf, acc);
  }
  int colg = q * K_ + nt * 16 + row;
  #pragma unroll
  for (int v = 0; v < 8; ++v) {
    int ch = mt * 16 + v + hi * 8;
    float y = acc[v];
    Y2[(size_t)ch * NCOL + colg] = y;
    stats_atomic(y, ch, row, s_sum, s_sq);
  }
}

// ---------------- conv3: BN2+ReLU fused load, 128x64 GEMM ----------------
// FINAL=false: accumulate BN3 stats only. FINAL=true: BN3+ReLU + max over K -> out.
template <bool FINAL>
__global__ void conv3_kernel(const float* __restrict__ Y2,
                             const float* __restrict__ W3,   // [128 x 64]
                             const float* __restrict__ a2, const float* __restrict__ b2p,
                             const float* __restrict__ a3, const float* __restrict__ b3p,
                             float* __restrict__ s_sum, float* __restrict__ s_sq,
                             float* __restrict__ out) {       // [B,128,M]
  __shared__ float X[64 * 32];
  __shared__ float redmax[256];
  int q = blockIdx.x;
  int b = q / M_;
  int m = q % M_;
  int t = threadIdx.x;
  int col = t & 31, part = t >> 5;
  for (int r = part; r < 64; r += 8) {
    float y = Y2[(size_t)r * NCOL + q * K_ + col];
    X[r * 32 + col] = fmaxf(a2[r] * y + b2p[r], 0.f);
  }
  __syncthreads();
  int lane = t & 31, wave = t >> 5;
  int nt = wave & 1, mt0 = wave >> 1;     // wave covers mt0 and mt0+4 (8 M-tiles total)
  int row = lane & 15, hi = lane >> 4;
  v8f acc0 = {0.f, 0.f, 0.f, 0.f, 0.f, 0.f, 0.f, 0.f};
  v8f acc1 = {0.f, 0.f, 0.f, 0.f, 0.f, 0.f, 0.f, 0.f};
  #pragma unroll
  for (int kk = 0; kk < 16; ++kk) {
    int c0 = kk * 4 + hi * 2;
    v2f bf;
    bf.x = X[c0 * 32 + nt * 16 + row];
    bf.y = X[(c0 + 1) * 32 + nt * 16 + row];
    v2f aA, aB;
    aA.x = W3[(mt0 * 16 + row) * 64 + c0];
    aA.y = W3[(mt0 * 16 + row) * 64 + c0 + 1];
    aB.x = W3[((mt0 + 4) * 16 + row) * 64 + c0];
    aB.y = W3[((mt0 + 4) * 16 + row) * 64 + c0 + 1];
    acc0 = wmma4(aA, bf, acc0);
    acc1 = wmma4(aB, bf, acc1);
  }
  #pragma unroll
  for (int half = 0; half < 2; ++half) {
    v8f acc = half ? acc1 : acc0;
    int mt = mt0 + half * 4;
    #pragma unroll
    for (int v = 0; v < 8; ++v) {
      int ch = mt * 16 + v + hi * 8;
      float y = acc[v];
      if (!FINAL) {
        stats_atomic(y, ch, row, s_sum, s_sq);
      } else {
        float h = fmaxf(a3[ch] * y + b3p[ch], 0.f);
        #pragma unroll
        for (int mask = 1; mask <= 8; mask <<= 1)
          h = fmaxf(h, __shfl_xor(h, mask));
        if (row == 0) redmax[ch * 2 + nt] = h;   // each (ch, nt) written once
      }
    }
  }
  if (FINAL) {
    __syncthreads();
    if (t < 128) {
      float r = fmaxf(redmax[t * 2], redmax[t * 2 + 1]);
      out[(size_t)b * 128 * M_ + (size_t)t * M_ + m] = r;
    }
  }
}

// ---------------- host launch ----------------
extern "C" void kernel_launch(void* const* d_in, const int* in_sizes, int n_in,
                              void* d_out, int out_size, void* d_ws, size_t ws_size,
                              hipStream_t stream) {
  (void)in_sizes; (void)n_in; (void)out_size; (void)ws_size;
  const float* loc     = (const float*)d_in[0];
  const float* new_loc = (const float*)d_in[1];
  const float* feat    = (const float*)d_in[2];
  const float* W1 = (const float*)d_in[3];
  const float* g1 = (const float*)d_in[4];
  const float* b1 = (const float*)d_in[5];
  const float* W2 = (const float*)d_in[6];
  const float* g2 = (const float*)d_in[7];
  const float* b2 = (const float*)d_in[8];
  const float* W3 = (const float*)d_in[9];
  const float* g3 = (const float*)d_in[10];
  const float* b3 = (const float*)d_in[11];
  float* out = (float*)d_out;

  char* ws = (char*)d_ws;
  float* stats  = (float*)ws;            // 1024 floats
  float* s1_sum = stats + 0;
  float* s1_sq  = stats + 64;
  float* s2_sum = stats + 128;
  float* s2_sq  = stats + 192;
  float* s3_sum = stats + 256;
  float* s3_sq  = stats + 384;
  float* a1  = stats + 512;
  float* b1p = stats + 576;
  float* a2  = stats + 640;
  float* b2p = stats + 704;
  float* a3  = stats + 768;
  float* b3p = stats + 896;
  size_t off = 4096;
  int*   idx   = (int*)(ws + off);                 off += (size_t)B_ * M_ * K_ * 4;     // 2 MB
  float* featT = (float*)(ws + off);               off += (size_t)B_ * N_ * 64 * 4;     // 16 MB
  float* Y1    = (float*)(ws + off);               off += (size_t)64 * NCOL * 4;        // 134 MB
  float* Y2    = (float*)(ws + off);

  hipMemsetAsync(stats, 0, 4096, stream);   // graph-capturable memset node

  feat_transpose_kernel<<<dim3(B_ * (N_ / 64)), 256, 0, stream>>>(feat, featT);
  knn_kernel<<<dim3(B_ * M_), 256, 0, stream>>>(loc, new_loc, idx);
  conv1_kernel<<<dim3(B_ * M_), 256, 0, stream>>>(idx, featT, loc, new_loc, W1, Y1,
                                                  s1_sum, s1_sq);
  finalize_bn<<<1, 128, 0, stream>>>(s1_sum, s1_sq, g1, b1, a1, b1p, 64);
  conv2_kernel<<<dim3(B_ * M_), 256, 0, stream>>>(Y1, W2, a1, b1p, Y2, s2_sum, s2_sq);
  finalize_bn<<<1, 128, 0, stream>>>(s2_sum, s2_sq, g2, b2, a2, b2p, 64);
  conv3_kernel<false><<<dim3(B_ * M_), 256, 0, stream>>>(Y2, W3, a2, b2p,
                                                         nullptr, nullptr,
                                                         s3_sum, s3_sq, nullptr);
  finalize_bn<<<1, 128, 0, stream>>>(s3_sum, s3_sq, g3, b3, a3, b3p, 128);
  conv3_kernel<true><<<dim3(B_ * M_), 256, 0, stream>>>(Y2, W3, a2, b2p,
                                                        a3, b3p,
                                                        nullptr, nullptr, out);
}